// NATBlock_21388937134562
// MI455X (gfx1250) — compile-verified
//
#include <hip/hip_runtime.h>
#include <math.h>

#define DIMC   128
#define NHEAD  4
#define HDIM   32
#define KW     7
#define NNB    49
#define HWDIM  56
#define BATCH  4
#define NTOK   (BATCH * HWDIM * HWDIM)   // 12544
#define MLPH   512

typedef float v2f __attribute__((ext_vector_type(2)));
typedef float v8f __attribute__((ext_vector_type(8)));

__device__ __forceinline__ float wave_sum(float v) {
#pragma unroll
    for (int o = 16; o > 0; o >>= 1) v += __shfl_xor(v, o, 32);
    return v;
}

// ---------------------------------------------------------------------------
// LayerNorm over last dim (128), one wave32 per token.
// lane loads float4 -> 32 lanes * 4 = 128 channels.
// ---------------------------------------------------------------------------
__global__ __launch_bounds__(256)
void ln_kernel(const float* __restrict__ x, const float* __restrict__ w,
               const float* __restrict__ b, float* __restrict__ out, int ntok) {
    int wave = (int)((blockIdx.x * blockDim.x + threadIdx.x) >> 5);
    int lane = threadIdx.x & 31;
    if (wave >= ntok) return;

    float4 v = ((const float4*)(x + (size_t)wave * DIMC))[lane];
    float s  = wave_sum(v.x + v.y + v.z + v.w);
    float sq = wave_sum(v.x * v.x + v.y * v.y + v.z * v.z + v.w * v.w);
    float mean = s * (1.0f / DIMC);
    float var  = sq * (1.0f / DIMC) - mean * mean;
    float rstd = rsqrtf(var + 1e-5f);

    float4 wv = ((const float4*)w)[lane];
    float4 bv = ((const float4*)b)[lane];
    float4 o;
    o.x = (v.x - mean) * rstd * wv.x + bv.x;
    o.y = (v.y - mean) * rstd * wv.y + bv.y;
    o.z = (v.z - mean) * rstd * wv.z + bv.z;
    o.w = (v.w - mean) * rstd * wv.w + bv.w;
    ((float4*)(out + (size_t)wave * DIMC))[lane] = o;
}

// ---------------------------------------------------------------------------
// f32 WMMA GEMM: out[M,N] = A[M,K] @ W[K,N] + bias (+ gelu) (+ residual)
// One wave32 per 16x16 output tile; K stepped by 4 via V_WMMA_F32_16X16X4_F32.
//
// A 16x4 fragment layout (ISA 7.12.2): lanes 0-15 hold M=lane, v0=K0,v1=K1;
// lanes 16-31 hold M=lane-16, v0=K2,v1=K3 -> one v2f (b64) load per lane.
// B 4x16 fragment mirrors it: v0 = row K0 (lo lanes)/K2 (hi), v1 = K1/K3.
// C/D: v[r] = row (r + 8*hiHalf), col = lane&15.
// ---------------------------------------------------------------------------
template <int ACT, bool HASRES>
__global__ __launch_bounds__(256)
void gemm16_wmma(const float* __restrict__ A, const float* __restrict__ W,
                 const float* __restrict__ bias, const float* __restrict__ res,
                 float* __restrict__ out, int M, int N, int K) {
    int wave = (int)((blockIdx.x * blockDim.x + threadIdx.x) >> 5);
    int lane = threadIdx.x & 31;
    int tilesN = N >> 4;
    int tm = wave / tilesN;
    int tn = wave - tm * tilesN;
    if (tm * 16 >= M) return;

    const bool hi = lane >= 16;
    const int  mr = lane & 15;

    const float* Arow = A + (size_t)(tm * 16 + mr) * K + (hi ? 2 : 0);
    const float* Bcol = W + (size_t)(hi ? 2 : 0) * N + tn * 16 + mr;

    __builtin_prefetch(Arow, 0, 3);
    __builtin_prefetch(Bcol, 0, 3);

    v8f acc = {};
#pragma unroll 8
    for (int k0 = 0; k0 < K; k0 += 4) {
        v2f a = *(const v2f*)(Arow + k0);          // K = k0+{0,1} (lo) / {2,3} (hi)
        v2f bb;
        bb.x = Bcol[(size_t)k0 * N];               // K = k0 + (hi?2:0)
        bb.y = Bcol[(size_t)(k0 + 1) * N];         // K = k0 + (hi?3:1)
        acc = __builtin_amdgcn_wmma_f32_16x16x4_f32(
            /*neg_a=*/false, a, /*neg_b=*/false, bb,
            /*c_mod=*/(short)0, acc, /*reuse_a=*/false, /*reuse_b=*/false);
    }

    int   col = tn * 16 + mr;
    float bv  = bias[col];
    int   row0 = tm * 16 + (hi ? 8 : 0);
#pragma unroll
    for (int r = 0; r < 8; ++r) {
        float v = acc[r] + bv;
        if (ACT == 1)  // exact GELU, matches jax.nn.gelu(approximate=False)
            v = 0.5f * v * (1.0f + erff(v * 0.70710678118654752f));
        size_t idx = (size_t)(row0 + r) * N + col;
        if (HASRES) v += res[idx];
        out[idx] = v;
    }
}

// ---------------------------------------------------------------------------
// Neighborhood attention: one wave32 per (token, head); lane = head dim.
// qkv layout per token: [q(4*32) | k(4*32) | v(4*32)], head-contiguous.
// Reference semantics: keys zero-padded (OOB logit = rpb only, OOB value = 0),
// softmax over all 49 positions.
// ---------------------------------------------------------------------------
__global__ __launch_bounds__(256)
void nat_attn(const float* __restrict__ qkv, const float* __restrict__ rpb,
              float* __restrict__ out) {
    int wave = (int)((blockIdx.x * blockDim.x + threadIdx.x) >> 5);
    int lane = threadIdx.x & 31;
    if (wave >= NTOK * NHEAD) return;

    int t = wave >> 2;
    int h = wave & 3;
    int bimg = t / (HWDIM * HWDIM);
    int rem  = t - bimg * (HWDIM * HWDIM);
    int y    = rem / HWDIM;
    int xw   = rem - y * HWDIM;

    const int hoff = h * HDIM + lane;
    float q = qkv[(size_t)t * 384 + hoff] * 0.17677669529663687f;  // 32^-0.5

    float logit[NNB];
    float mx = -1e30f;
    for (int ky = 0; ky < KW; ++ky) {
        int yy = y + ky - 3;
        for (int kx = 0; kx < KW; ++kx) {
            int xx = xw + kx - 3;
            int j  = ky * KW + kx;
            float kvv = 0.0f;
            if (yy >= 0 && yy < HWDIM && xx >= 0 && xx < HWDIM) {
                int tnb = (bimg * HWDIM + yy) * HWDIM + xx;
                kvv = qkv[(size_t)tnb * 384 + 128 + hoff];
            }
            float d = wave_sum(q * kvv) + rpb[h * NNB + j];
            logit[j] = d;
            mx = fmaxf(mx, d);
        }
    }

    float ssum = 0.0f;
#pragma unroll
    for (int j = 0; j < NNB; ++j) {
        float e = expf(logit[j] - mx);
        logit[j] = e;
        ssum += e;
    }
    float inv = 1.0f / ssum;

    float o = 0.0f;
    for (int ky = 0; ky < KW; ++ky) {
        int yy = y + ky - 3;
        for (int kx = 0; kx < KW; ++kx) {
            int xx = xw + kx - 3;
            if (yy >= 0 && yy < HWDIM && xx >= 0 && xx < HWDIM) {
                int tnb = (bimg * HWDIM + yy) * HWDIM + xx;
                o += logit[ky * KW + kx] * qkv[(size_t)tnb * 384 + 256 + hoff];
            }
        }
    }
    out[(size_t)t * DIMC + hoff] = o * inv;
}

// ---------------------------------------------------------------------------
extern "C" void kernel_launch(void* const* d_in, const int* in_sizes, int n_in,
                              void* d_out, int out_size, void* d_ws, size_t ws_size,
                              hipStream_t stream) {
    (void)in_sizes; (void)n_in; (void)out_size; (void)ws_size;

    const float* x     = (const float*)d_in[0];
    const float* ln1w  = (const float*)d_in[1];
    const float* ln1b  = (const float*)d_in[2];
    const float* wqkv  = (const float*)d_in[3];
    const float* bqkv  = (const float*)d_in[4];
    const float* rpb   = (const float*)d_in[5];
    const float* wproj = (const float*)d_in[6];
    const float* bproj = (const float*)d_in[7];
    const float* ln2w  = (const float*)d_in[8];
    const float* ln2b  = (const float*)d_in[9];
    const float* wfc1  = (const float*)d_in[10];
    const float* bfc1  = (const float*)d_in[11];
    const float* wfc2  = (const float*)d_in[12];
    const float* bfc2  = (const float*)d_in[13];
    float* out = (float*)d_out;

    float* ws   = (float*)d_ws;
    float* xn   = ws;                                // N*128
    float* qkv  = xn   + (size_t)NTOK * DIMC;        // N*384
    float* attn = qkv  + (size_t)NTOK * 384;         // N*128
    float* x1   = attn + (size_t)NTOK * DIMC;        // N*128
    float* xn2  = x1   + (size_t)NTOK * DIMC;        // N*128
    float* hbuf = xn2  + (size_t)NTOK * DIMC;        // N*512

    const dim3 blk(256);
    auto blocks_for_waves = [](long waves) { return (unsigned)((waves * 32 + 255) / 256); };

    // 1) LN1
    ln_kernel<<<blocks_for_waves(NTOK), blk, 0, stream>>>(x, ln1w, ln1b, xn, NTOK);

    // 2) QKV = xn @ w_qkv + b_qkv       [12544 x 384], K=128
    long tiles = (long)(NTOK / 16) * (384 / 16);
    gemm16_wmma<0, false><<<blocks_for_waves(tiles), blk, 0, stream>>>(
        xn, wqkv, bqkv, nullptr, qkv, NTOK, 384, DIMC);

    // 3) Neighborhood attention
    nat_attn<<<blocks_for_waves((long)NTOK * NHEAD), blk, 0, stream>>>(qkv, rpb, attn);

    // 4) x1 = x + attn @ w_proj + b_proj    [12544 x 128], K=128
    tiles = (long)(NTOK / 16) * (DIMC / 16);
    gemm16_wmma<0, true><<<blocks_for_waves(tiles), blk, 0, stream>>>(
        attn, wproj, bproj, x, x1, NTOK, DIMC, DIMC);

    // 5) LN2
    ln_kernel<<<blocks_for_waves(NTOK), blk, 0, stream>>>(x1, ln2w, ln2b, xn2, NTOK);

    // 6) h = gelu(xn2 @ w_fc1 + b_fc1)      [12544 x 512], K=128
    tiles = (long)(NTOK / 16) * (MLPH / 16);
    gemm16_wmma<1, false><<<blocks_for_waves(tiles), blk, 0, stream>>>(
        xn2, wfc1, bfc1, nullptr, hbuf, NTOK, MLPH, DIMC);

    // 7) out = x1 + h @ w_fc2 + b_fc2       [12544 x 128], K=512
    tiles = (long)(NTOK / 16) * (DIMC / 16);
    gemm16_wmma<0, true><<<blocks_for_waves(tiles), blk, 0, stream>>>(
        hbuf, wfc2, bfc2, x1, out, NTOK, DIMC, MLPH);
}